// AttnPool_45114336477919
// MI455X (gfx1250) — compile-verified
//
#include <hip/hip_runtime.h>
#include <cmath>

// ---------------------------------------------------------------------------
// AttnPool on MI455X (gfx1250, wave32).
//
// Algebraic rewrite: scores[b,p] = z[b,p] . (W^T q) * 1/sqrt(D), so the big
// [B,P,D]x[D,D] GEMM disappears. The kernel is HBM-bound: two streaming
// passes over z (2 x 201 MB -> ~17us at 23.3 TB/s; pass 2 largely L2-resident
// since |z| ~= 192 MB L2). Dot products in pass 1 ride the matrix pipe via
// V_WMMA_F32_16X16X4_F32 with a broadcast B operand (all 16 columns = qp
// chunk), which is free compute under a bandwidth roofline.
// ---------------------------------------------------------------------------

typedef __attribute__((ext_vector_type(2))) float v2f;
typedef __attribute__((ext_vector_type(8))) float v8f;

#define DIM   768
#define ROWS  1024
#define BATCH 64

// K0: qp[d] = (1/sqrt(D)) * sum_e q[e] * W[e,d].  Coalesced over d.
__global__ __launch_bounds__(128, 1)
void qproj_kernel(const float* __restrict__ q, const float* __restrict__ W,
                  float* __restrict__ qp) {
    __shared__ float qs[DIM];
    for (int i = threadIdx.x; i < DIM; i += 128) qs[i] = q[i];
    __syncthreads();
    const int d = blockIdx.x * 128 + threadIdx.x;   // grid 6 x 128 == 768
    float acc = 0.f;
    for (int e = 0; e < DIM; ++e) acc = fmaf(qs[e], W[e * DIM + d], acc);
    qp[d] = acc * 0.03608439182435161f;             // 1/sqrt(768)
}

// K1: raw scores via WMMA.  Block = (row-chunk, batch), 8 waves, each wave
// computes 2 tiles of 16 rows.  A fragment: lane(l) = row p0 + (l&15),
// K = {0,1} for lanes 0-15, {2,3} for lanes 16-31 (ISA 16x4 f32 A layout).
// B fragment: every column = qp chunk, so lane half selects the same K pair.
// D: VGPR j holds rows p0+j (lanes 0-15) / p0+8+j (lanes 16-31); we read col 0.
__global__ __launch_bounds__(256, 1)
void scores_kernel(const float* __restrict__ z, const float* __restrict__ qp,
                   float* __restrict__ sraw) {
    __shared__ float qs[DIM];
    const int tid = threadIdx.x;
    for (int i = tid; i < DIM; i += 256) qs[i] = qp[i];
    __syncthreads();

    const int b     = blockIdx.y;
    const int chunk = blockIdx.x;                  // 256-row chunk, 0..3
    const float* zb = z + ((size_t)b * ROWS + (size_t)chunk * 256) * DIM;

    const int wave = tid >> 5;
    const int lane = tid & 31;
    const int half = lane >> 4;                    // 0: K=0,1   1: K=2,3
    const int m    = lane & 15;                    // row within tile

    for (int t = wave; t < 16; t += 8) {           // 16 tiles, uniform per wave
        const int p0 = t * 16;
        const float* rowp = zb + (size_t)(p0 + m) * DIM + 2 * half;
        v8f acc = {};
        for (int k0 = 0; k0 < DIM; k0 += 4) {
            v2f a = *(const v2f*)(rowp + k0);      // A[m][k0+2h .. +1]
            v2f bb;
            bb.x = qs[k0 + 2 * half];              // B[k][n] = qp[k0+k]  (all n)
            bb.y = qs[k0 + 2 * half + 1];
            acc = __builtin_amdgcn_wmma_f32_16x16x4_f32(
                false, a, false, bb, (short)0, acc, false, false);
        }
        if (m == 0) {                              // lane 0 -> rows p0..p0+7,
            float* sp = sraw + (size_t)b * ROWS    // lane 16 -> rows p0+8..+15
                        + (size_t)chunk * 256 + p0 + half * 8;
#pragma unroll
            for (int j = 0; j < 8; ++j) sp[j] = acc[j];
        }
    }
}

// K2: softmax over the 1024 scores of one batch, in place -> weights.
__global__ __launch_bounds__(256, 1)
void softmax_kernel(float* __restrict__ s) {
    __shared__ float redm[8];
    __shared__ float reds[8];
    const int tid = threadIdx.x;
    float* sb = s + (size_t)blockIdx.x * ROWS;

    float v0 = sb[tid], v1 = sb[tid + 256], v2 = sb[tid + 512], v3 = sb[tid + 768];
    float mx = fmaxf(fmaxf(v0, v1), fmaxf(v2, v3));
#pragma unroll
    for (int off = 16; off > 0; off >>= 1) mx = fmaxf(mx, __shfl_down(mx, off));
    if ((tid & 31) == 0) redm[tid >> 5] = mx;
    __syncthreads();
    mx = redm[0];
#pragma unroll
    for (int i = 1; i < 8; ++i) mx = fmaxf(mx, redm[i]);

    float e0 = __expf(v0 - mx), e1 = __expf(v1 - mx);
    float e2 = __expf(v2 - mx), e3 = __expf(v3 - mx);
    float ps = (e0 + e1) + (e2 + e3);
#pragma unroll
    for (int off = 16; off > 0; off >>= 1) ps += __shfl_down(ps, off);
    if ((tid & 31) == 0) reds[tid >> 5] = ps;
    __syncthreads();
    float tot = 0.f;
#pragma unroll
    for (int i = 0; i < 8; ++i) tot += reds[i];
    const float inv = 1.0f / tot;

    sb[tid]       = e0 * inv;
    sb[tid + 256] = e1 * inv;
    sb[tid + 512] = e2 * inv;
    sb[tid + 768] = e3 * inv;
}

// K3: pooled[b,d] = sum_p w[b,p] * z[b,p,d].  Block = (col-chunk, batch),
// thread owns one d column; loads are lane-coalesced at each p; weights
// broadcast from LDS.  256 independent blocks -> saturates HBM.
__global__ __launch_bounds__(192, 1)
void pool_kernel(const float* __restrict__ z, const float* __restrict__ w,
                 float* __restrict__ out) {
    __shared__ float ws[ROWS];
    const int tid = threadIdx.x;
    const int b   = blockIdx.y;
    for (int i = tid; i < ROWS; i += 192) ws[i] = w[(size_t)b * ROWS + i];
    __syncthreads();

    const int d = blockIdx.x * 192 + tid;          // grid.x = 4 -> 768 cols
    const float* col = z + (size_t)b * ROWS * DIM + d;
    float a0 = 0.f, a1 = 0.f, a2 = 0.f, a3 = 0.f;
    for (int p = 0; p < ROWS; p += 4) {
        a0 = fmaf(ws[p + 0], col[(size_t)(p + 0) * DIM], a0);
        a1 = fmaf(ws[p + 1], col[(size_t)(p + 1) * DIM], a1);
        a2 = fmaf(ws[p + 2], col[(size_t)(p + 2) * DIM], a2);
        a3 = fmaf(ws[p + 3], col[(size_t)(p + 3) * DIM], a3);
    }
    out[(size_t)b * DIM + d] = (a0 + a1) + (a2 + a3);
}

extern "C" void kernel_launch(void* const* d_in, const int* in_sizes, int n_in,
                              void* d_out, int out_size, void* d_ws, size_t ws_size,
                              hipStream_t stream) {
    (void)in_sizes; (void)n_in; (void)out_size; (void)ws_size;
    const float* z = (const float*)d_in[0];   // [64,1024,768] f32
    const float* q = (const float*)d_in[1];   // [768] f32
    const float* W = (const float*)d_in[2];   // [768,768] f32
    float* out  = (float*)d_out;              // [64,768] f32
    float* qp   = (float*)d_ws;               // 768 floats
    float* sraw = qp + DIM;                   // 64*1024 floats (scores->weights)

    qproj_kernel  <<<dim3(6),      dim3(128), 0, stream>>>(q, W, qp);
    scores_kernel <<<dim3(4, 64),  dim3(256), 0, stream>>>(z, qp, sraw);
    softmax_kernel<<<dim3(64),     dim3(256), 0, stream>>>(sraw);
    pool_kernel   <<<dim3(4, 64),  dim3(192), 0, stream>>>(z, sraw, out);
}